// AttHGCNConv_9740985828005
// MI455X (gfx1250) — compile-verified
//
#include <hip/hip_runtime.h>

typedef __attribute__((ext_vector_type(2))) float v2f;
typedef __attribute__((ext_vector_type(8))) float v8f;

#define TILE_M 128
#define TILE_N 64
#define TILE_K 32
#define A_STRIDE 36   // 144 B/row: 16B-aligned float4 stores, gcd(36,64)=4 -> good bank spread
#define X_STRIDE 72   // 288 B/row: 16B-aligned, k-row bank shift of 8 avoids half-wave conflicts

// C[M x D] = bigOp[M x K] @ X[K x D], bigOp = TRANS ? big^T : big
// big is stored with leading dimension ldBig; TRANS reads big[k*ldBig + m].
template <bool TRANS, bool LEAKY>
__global__ __launch_bounds__(256)
void gemm_skinny_f32(const float* __restrict__ big, const float* __restrict__ X,
                     float* __restrict__ Y, int K, int ldBig, int D) {
  __shared__ float As[2][TILE_M * A_STRIDE];
  __shared__ float Xs[2][TILE_K * X_STRIDE];

  const int tid  = threadIdx.x;
  const int m0   = blockIdx.x * TILE_M;
  const int n0   = blockIdx.y * TILE_N;
  const int wave = tid >> 5;
  const int lane = tid & 31;
  const int half = lane >> 4;   // K/M half-select per ISA layout
  const int l    = lane & 15;
  const int wm   = wave >> 1;   // 0..3  (M sub-tile of 32)
  const int wn   = wave & 1;    // 0..1  (N sub-tile of 32)

  v8f acc[2][2];
#pragma unroll
  for (int mi = 0; mi < 2; ++mi)
#pragma unroll
    for (int ni = 0; ni < 2; ++ni)
      acc[mi][ni] = v8f{0.f, 0.f, 0.f, 0.f, 0.f, 0.f, 0.f, 0.f};

  float4 aReg[4];
  float4 xReg[2];

  // ---- global -> registers (prefetch) ----
  auto fetch = [&](int k0) {
    if constexpr (!TRANS) {
#pragma unroll
      for (int p = 0; p < 4; ++p) {
        int q   = p * 256 + tid;       // 0..1023
        int row = q >> 3;              // 0..127 (m)
        int jv  = (q & 7) << 2;        // 0..28  (k, float4)
        aReg[p] = *reinterpret_cast<const float4*>(
            &big[(size_t)(m0 + row) * ldBig + (size_t)k0 + jv]);
      }
    } else {
#pragma unroll
      for (int p = 0; p < 4; ++p) {
        int q  = p * 256 + tid;        // 0..1023
        int j  = q >> 5;               // 0..31  (k row in big)
        int iv = (q & 31) << 2;        // 0..124 (m, float4, contiguous)
        aReg[p] = *reinterpret_cast<const float4*>(
            &big[(size_t)(k0 + j) * ldBig + (size_t)m0 + iv]);
      }
    }
#pragma unroll
    for (int p = 0; p < 2; ++p) {
      int q  = p * 256 + tid;          // 0..511
      int j  = q >> 4;                 // 0..31 (k)
      int nv = (q & 15) << 2;          // 0..60 (n, float4)
      xReg[p] = *reinterpret_cast<const float4*>(
          &X[(size_t)(k0 + j) * D + (size_t)n0 + nv]);
    }
  };

  // ---- registers -> LDS ----
  auto stage = [&](int buf) {
    float* as = As[buf];
    float* xs = Xs[buf];
    if constexpr (!TRANS) {
#pragma unroll
      for (int p = 0; p < 4; ++p) {
        int q   = p * 256 + tid;
        int row = q >> 3;
        int jv  = (q & 7) << 2;
        *reinterpret_cast<float4*>(&as[row * A_STRIDE + jv]) = aReg[p];
      }
    } else {
#pragma unroll
      for (int p = 0; p < 4; ++p) {
        int q  = p * 256 + tid;
        int j  = q >> 5;
        int iv = (q & 31) << 2;
        as[(iv + 0) * A_STRIDE + j] = aReg[p].x;  // transpose on the way in
        as[(iv + 1) * A_STRIDE + j] = aReg[p].y;
        as[(iv + 2) * A_STRIDE + j] = aReg[p].z;
        as[(iv + 3) * A_STRIDE + j] = aReg[p].w;
      }
    }
#pragma unroll
    for (int p = 0; p < 2; ++p) {
      int q  = p * 256 + tid;
      int j  = q >> 4;
      int nv = (q & 15) << 2;
      *reinterpret_cast<float4*>(&xs[j * X_STRIDE + nv]) = xReg[p];
    }
  };

  // ---- WMMA over one K-tile from LDS ----
  // A-frag (16x4 f32): lane<16 -> (K=0,K=1) of row M=l ; lane>=16 -> (K=2,K=3)
  // B-frag (4x16 f32): mirrored K split across lane halves, N=l across lanes
  auto compute = [&](int buf) {
    const float* as = As[buf];
    const float* xs = Xs[buf];
#pragma unroll
    for (int kk = 0; kk < TILE_K / 4; ++kk) {
      const int kb = kk * 4 + half * 2;
      v2f a[2], b[2];
#pragma unroll
      for (int mi = 0; mi < 2; ++mi) {
        const float* ap = &as[(wm * 32 + mi * 16 + l) * A_STRIDE + kb];
        a[mi].x = ap[0];
        a[mi].y = ap[1];
      }
#pragma unroll
      for (int ni = 0; ni < 2; ++ni) {
        const int col = wn * 32 + ni * 16 + l;
        b[ni].x = xs[(kb + 0) * X_STRIDE + col];
        b[ni].y = xs[(kb + 1) * X_STRIDE + col];
      }
#pragma unroll
      for (int mi = 0; mi < 2; ++mi)
#pragma unroll
        for (int ni = 0; ni < 2; ++ni)
          acc[mi][ni] = __builtin_amdgcn_wmma_f32_16x16x4_f32(
              false, a[mi], false, b[ni], (short)0, acc[mi][ni], false, false);
    }
  };

  const int nkt = K / TILE_K;
  fetch(0);
  stage(0);
  for (int kt = 0; kt < nkt; ++kt) {
    if (kt + 1 < nkt) fetch((kt + 1) * TILE_K);  // overlap global latency with WMMA
    __syncthreads();                             // stage(kt) visible; prev compute done
    compute(kt & 1);
    if (kt + 1 < nkt) stage((kt + 1) & 1);       // other buffer: safe after barrier
  }

  // ---- epilogue: C/D 16x16 f32 layout: VGPR r -> M = half*8 + r, N = l ----
#pragma unroll
  for (int mi = 0; mi < 2; ++mi)
#pragma unroll
    for (int ni = 0; ni < 2; ++ni)
#pragma unroll
      for (int r = 0; r < 8; ++r) {
        int row = m0 + wm * 32 + mi * 16 + half * 8 + r;
        int col = n0 + wn * 32 + ni * 16 + l;
        float v = acc[mi][ni][r];
        if constexpr (LEAKY) v = v > 0.f ? v : 0.2f * v;
        Y[(size_t)row * D + col] = v;
      }
}

extern "C" void kernel_launch(void* const* d_in, const int* in_sizes, int n_in,
                              void* d_out, int out_size, void* d_ws, size_t ws_size,
                              hipStream_t stream) {
  (void)in_sizes; (void)n_in; (void)out_size; (void)ws_size;
  constexpr int N = 8192, E = 8192, D = 256;

  const float* inp_adj = (const float*)d_in[0];  // (E, N)
  const float* att_adj = (const float*)d_in[1];  // (N, E)
  const float* embs    = (const float*)d_in[2];  // (N, D)
  float* out = (float*)d_out;                    // (N, D)
  float* ws  = (float*)d_ws;                     // 8 MB scratch used

  dim3 block(256);
  dim3 grid(8192 / TILE_M, D / TILE_N);  // all four GEMMs: M=8192, N=256

  // t1 = att_adj^T @ embs        (E x D) -> ws
  gemm_skinny_f32<true, false><<<grid, block, 0, stream>>>(att_adj, embs, ws, N, E, D);
  // t2 = inp_adj^T @ t1          (N x D) -> d_out (scratch; overwritten by call 4)
  gemm_skinny_f32<true, false><<<grid, block, 0, stream>>>(inp_adj, ws, out, E, N, D);
  // t3 = inp_adj @ t2            (E x D) -> ws
  gemm_skinny_f32<false, false><<<grid, block, 0, stream>>>(inp_adj, out, ws, N, N, D);
  // out = LeakyReLU(att_adj @ t3) (N x D) -> d_out
  gemm_skinny_f32<false, true><<<grid, block, 0, stream>>>(att_adj, ws, out, E, E, D);
}